// AdaptiveSparsityLayer_88029649699387
// MI455X (gfx1250) — compile-verified
//
#include <hip/hip_runtime.h>

#define FEATURES 4096
#define BATCH 16384
#define KSEL 409
#define NT 256
#define VPT 4  // float4 chunks per thread: 4 * 256 * 4 = 4096 floats

typedef __attribute__((ext_vector_type(2))) float v2f;
typedef __attribute__((ext_vector_type(8))) float v8f;

// Match the builtin's parameter pointee type exactly: GCC-style int vector, 16 bytes.
typedef int v4si __attribute__((vector_size(16)));
typedef __attribute__((address_space(1))) v4si gv4;  // global (device) AS
typedef __attribute__((address_space(3))) v4si lv4;  // LDS AS

#if __has_builtin(__builtin_amdgcn_global_load_async_to_lds_b128)
#define USE_ASYNC 1
#pragma message("CDNA5: async global->LDS builtin AVAILABLE (using it)")
#else
#define USE_ASYNC 0
#pragma message("CDNA5: async global->LDS builtin NOT available; plain copy fallback")
#endif

#if __has_builtin(__builtin_amdgcn_wmma_f32_16x16x4_f32)
#define USE_WMMA 1
#pragma message("CDNA5: wmma_f32_16x16x4_f32 builtin AVAILABLE (using it)")
#else
#define USE_WMMA 0
#pragma message("CDNA5: wmma_f32_16x16x4_f32 builtin NOT available; scalar reduce fallback")
#endif

__device__ __forceinline__ unsigned ordkey(float f) {
  unsigned u = __float_as_uint(f);
  // monotonic float -> uint mapping (larger float => larger key)
  return (u & 0x80000000u) ? ~u : (u | 0x80000000u);
}

__global__ __launch_bounds__(NT) void ln_topk_kernel(const float* __restrict__ x,
                                                     const float* __restrict__ gamma,
                                                     const float* __restrict__ beta,
                                                     float* __restrict__ out) {
  __shared__ __align__(16) float xnS[FEATURES];  // 16 KB: raw row, then normalized row
  __shared__ unsigned histS[NT];                 // radix histogram / scan buffer
  __shared__ float wredS[16];                    // [0..7] wave sums, [8..15] wave sumsqs
  __shared__ float red2S[16];                    // WMMA column sums
  __shared__ float statS[2];                     // mean, rstd
  __shared__ unsigned selS[4];                   // digit, remaining, eqCount, tieBase

  const int tid = threadIdx.x;
  const int wid = tid >> 5;
  const int lane = tid & 31;
  const size_t row = blockIdx.x;
  const float* xrow = x + row * (size_t)FEATURES;
  float* orow = out + row * (size_t)FEATURES;

  float4 r[VPT];

  // ---------------- Phase 0: stream row into LDS (async-tensor path) -------------
#if USE_ASYNC
#pragma unroll
  for (int c = 0; c < VPT; ++c) {
    const int f4 = c * NT + tid;
    gv4* gp = (gv4*)(size_t)(xrow + f4 * 4);              // int -> AS1 ptr (same VA)
    lv4* lp = (lv4*)(unsigned)(size_t)(&xnS[f4 * 4]);     // low 32 bits = LDS offset
    __builtin_amdgcn_global_load_async_to_lds_b128(gp, lp, 0, 0);
  }
#if __has_builtin(__builtin_amdgcn_s_wait_asynccnt)
  __builtin_amdgcn_s_wait_asynccnt(0);
#else
  asm volatile("s_wait_asynccnt 0" ::: "memory");
#endif
  __syncthreads();
#pragma unroll
  for (int c = 0; c < VPT; ++c) r[c] = ((const float4*)xnS)[c * NT + tid];
#else
#pragma unroll
  for (int c = 0; c < VPT; ++c) {
    r[c] = ((const float4*)xrow)[c * NT + tid];
    ((float4*)xnS)[c * NT + tid] = r[c];
  }
  __syncthreads();
#endif

  // ---------------- Phase 1: mean / variance ------------------------------------
  float s = 0.f, q = 0.f;
#pragma unroll
  for (int c = 0; c < VPT; ++c) {
    float4 v = r[c];
    s += v.x + v.y + v.z + v.w;
    q += v.x * v.x + v.y * v.y + v.z * v.z + v.w * v.w;
  }
#pragma unroll
  for (int o = 16; o > 0; o >>= 1) {
    s += __shfl_xor(s, o, 32);
    q += __shfl_xor(q, o, 32);
  }
  if (lane == 0) { wredS[wid] = s; wredS[8 + wid] = q; }
  __syncthreads();

#if USE_WMMA
  // Cross-wave combine on the matrix pipe: D = ones(16x4) x B, so D[0,n] = sum_k B[k,n].
  // Column 0/1 carry the 8 wave-sums, column 2/3 the 8 wave-sumsqs. The result only
  // depends on the SET of 4 values per column, so it is robust to the exact K<->VGPR map.
  if (wid == 0) {
    const int n = lane & 15;
    float b0 = 0.f, b1 = 0.f;
    if (n < 4) {
      const int base = n * 4;                 // n=0: ws[0..3], n=1: ws[4..7], n=2/3: wq
      const int k0 = (lane < 16) ? 0 : 2;
      b0 = wredS[base + k0];
      b1 = wredS[base + k0 + 1];
    }
    v2f A; A[0] = 1.0f; A[1] = 1.0f;
    v2f B; B[0] = b0;   B[1] = b1;
    v8f C = {0.f, 0.f, 0.f, 0.f, 0.f, 0.f, 0.f, 0.f};
    v8f D = __builtin_amdgcn_wmma_f32_16x16x4_f32(false, A, false, B, (short)0, C, false, false);
    if (lane < 16) red2S[lane] = D[0];        // row M=0 lives in VGPR0, lanes 0..15
  }
  __syncthreads();
#else
  if (tid == 0) {
    float ts = 0.f, tq = 0.f;
    for (int i = 0; i < 8; ++i) { ts += wredS[i]; tq += wredS[8 + i]; }
    red2S[0] = ts; red2S[1] = 0.f; red2S[2] = tq; red2S[3] = 0.f;
  }
  __syncthreads();
#endif

  if (tid == 0) {
    const float sum = red2S[0] + red2S[1];
    const float ssq = red2S[2] + red2S[3];
    const float mean = sum * (1.0f / FEATURES);
    const float var = ssq * (1.0f / FEATURES) - mean * mean;
    statS[0] = mean;
    statS[1] = rsqrtf(var + 1e-5f);
  }
  __syncthreads();
  const float mean = statS[0], rstd = statS[1];

  // ---------------- Phase 2: normalize into LDS ---------------------------------
#pragma unroll
  for (int c = 0; c < VPT; ++c) {
    const int f4 = c * NT + tid;
    const float4 g = ((const float4*)gamma)[f4];
    const float4 b = ((const float4*)beta)[f4];
    const float4 v = r[c];
    float4 y;
    y.x = (v.x - mean) * rstd * g.x + b.x;
    y.y = (v.y - mean) * rstd * g.y + b.y;
    y.z = (v.z - mean) * rstd * g.z + b.z;
    y.w = (v.w - mean) * rstd * g.w + b.w;
    ((float4*)xnS)[f4] = y;
  }
  __syncthreads();

  // ---------------- Phase 3: exact K-th-largest via 4x8-bit radix select --------
  unsigned remaining = KSEL;
  unsigned maskHigh = 0u, prefixVal = 0u;
#pragma unroll 1
  for (int pass = 0; pass < 4; ++pass) {
    const int shift = 24 - 8 * pass;
    histS[tid] = 0u;
    __syncthreads();
#pragma unroll
    for (int c = 0; c < VPT; ++c) {
      const float4 v = ((const float4*)xnS)[c * NT + tid];
      const float e[4] = {v.x, v.y, v.z, v.w};
#pragma unroll
      for (int j = 0; j < 4; ++j) {
        const unsigned key = ordkey(e[j]);
        if ((key & maskHigh) == prefixVal)
          atomicAdd(&histS[(key >> shift) & 255u], 1u);
      }
    }
    __syncthreads();
    // inclusive suffix-sum (count of candidates with digit >= tid)
#pragma unroll 1
    for (int off = 1; off < NT; off <<= 1) {
      const unsigned a = histS[tid];
      const unsigned b = (tid + off < NT) ? histS[tid + off] : 0u;
      __syncthreads();
      histS[tid] = a + b;
      __syncthreads();
    }
    const unsigned sfx = histS[tid];
    const unsigned nxt = (tid < NT - 1) ? histS[tid + 1] : 0u;
    if (sfx >= remaining && (tid == NT - 1 || nxt < remaining)) {
      selS[0] = (unsigned)tid;         // selected digit
      selS[1] = remaining - nxt;       // remaining among this digit
      selS[2] = sfx - nxt;             // count with exactly this digit
    }
    __syncthreads();
    prefixVal |= (selS[0] << shift);
    maskHigh |= (0xFFu << shift);
    remaining = selS[1];
    __syncthreads();                   // protect histS / selS before next pass
  }
  const unsigned T = prefixVal;        // key of K-th largest element
  const unsigned nneed = remaining;    // how many key==T to keep
  const unsigned eqTotal = selS[2];    // how many key==T exist

  // ---------------- Phase 4: masked write-out -----------------------------------
  if (eqTotal == nneed) {
    // no contested ties: keep everything with key >= T
#pragma unroll
    for (int c = 0; c < VPT; ++c) {
      const int f4 = c * NT + tid;
      const float4 v = ((const float4*)xnS)[f4];
      float4 o;
      o.x = (ordkey(v.x) >= T) ? v.x : 0.f;
      o.y = (ordkey(v.y) >= T) ? v.y : 0.f;
      o.z = (ordkey(v.z) >= T) ? v.z : 0.f;
      o.w = (ordkey(v.w) >= T) ? v.w : 0.f;
      ((float4*)orow)[f4] = o;
    }
  } else {
    // ties at threshold: keep the lowest-index nneed of them (matches jax top_k)
    if (tid == 0) selS[3] = 0u;
    __syncthreads();
#pragma unroll 1
    for (int c = 0; c < VPT; ++c) {
      const int f4 = c * NT + tid;
      const float4 v = ((const float4*)xnS)[f4];
      const float e[4] = {v.x, v.y, v.z, v.w};
      unsigned k[4], cnt = 0;
#pragma unroll
      for (int j = 0; j < 4; ++j) { k[j] = ordkey(e[j]); cnt += (k[j] == T) ? 1u : 0u; }
      histS[tid] = cnt;
      __syncthreads();
#pragma unroll 1
      for (int off = 1; off < NT; off <<= 1) {   // ascending inclusive scan
        const unsigned a = histS[tid];
        const unsigned b = (tid >= off) ? histS[tid - off] : 0u;
        __syncthreads();
        histS[tid] = a + b;
        __syncthreads();
      }
      const unsigned incl = histS[tid];
      const unsigned total = histS[NT - 1];
      unsigned rank = selS[3] + (incl - cnt);    // global index-order rank among equals
      float o[4];
#pragma unroll
      for (int j = 0; j < 4; ++j) {
        if (k[j] > T) o[j] = e[j];
        else if (k[j] == T) { o[j] = (rank < nneed) ? e[j] : 0.f; ++rank; }
        else o[j] = 0.f;
      }
      float4 ov; ov.x = o[0]; ov.y = o[1]; ov.z = o[2]; ov.w = o[3];
      ((float4*)orow)[f4] = ov;
      __syncthreads();
      if (tid == 0) selS[3] += total;
      __syncthreads();
    }
  }
}

extern "C" void kernel_launch(void* const* d_in, const int* in_sizes, int n_in,
                              void* d_out, int out_size, void* d_ws, size_t ws_size,
                              hipStream_t stream) {
  (void)in_sizes; (void)n_in; (void)out_size; (void)d_ws; (void)ws_size;
  const float* x = (const float*)d_in[0];
  const float* gamma = (const float*)d_in[1];
  const float* beta = (const float*)d_in[2];
  float* out = (float*)d_out;
  ln_topk_kernel<<<dim3(BATCH), dim3(NT), 0, stream>>>(x, gamma, beta, out);
}